// KGAT_52441550684532
// MI455X (gfx1250) — compile-verified
//
#include <hip/hip_runtime.h>
#include <hip/hip_bf16.h>

#define N_NODES 150000
#define N_EDGES 1200000
#define DIM 64

typedef __attribute__((ext_vector_type(2))) float v2f;
typedef __attribute__((ext_vector_type(8))) float v8f;

// ---------- helpers ----------
__device__ __forceinline__ unsigned f2ord(float f) {
    unsigned u = __float_as_uint(f);
    return (u & 0x80000000u) ? ~u : (u | 0x80000000u);
}
__device__ __forceinline__ float ord2f(unsigned v) {
    unsigned u = (v & 0x80000000u) ? (v ^ 0x80000000u) : ~v;
    return __uint_as_float(u);
}

// ---------- init kernels ----------
__global__ void k_init_xacc(const float* __restrict__ emb, float* __restrict__ x,
                            float* __restrict__ acc, int n) {
    int i = blockIdx.x * blockDim.x + threadIdx.x;
    if (i < n) { float v = emb[i]; x[i] = v; acc[i] = v; }
}

__global__ void k_zero(float* __restrict__ p, int n) {
    int i = blockIdx.x * blockDim.x + threadIdx.x;
    if (i < n) p[i] = 0.0f;
}

__global__ void k_init_max(unsigned* maxbits) {
    *maxbits = f2ord(-3.0e38f);
}

// ---------- GEMM: xt = x @ W (row-major, K=N=64) via V_WMMA_F32_16X16X4_F32 ----------
__global__ __launch_bounds__(256) void k_gemm_xw(const float* __restrict__ x,
                                                 const float* __restrict__ W,
                                                 float* __restrict__ xt, int n_rows) {
    __shared__ float Wl[DIM * DIM];  // 16 KB
    for (int i = threadIdx.x; i < DIM * DIM / 4; i += 256)
        ((float4*)Wl)[i] = ((const float4*)W)[i];
    __syncthreads();

    const int wave = threadIdx.x >> 5;     // 0..7
    const int lane = threadIdx.x & 31;
    const int half = lane >> 4;            // 0|1 -> K pair / M+8 select
    const int l16  = lane & 15;

    const int row0 = (blockIdx.x * 8 + wave) * 16;   // wave-uniform
    if (row0 >= n_rows) return;                      // whole wave exits: EXEC stays full

    // Preload all 16 A fragments for this 16x64 strip.
    // A layout (ISA 7.12.2, 32-bit A 16x4): lane l16 holds row M=row0+l16,
    // v0=K base+2*half, v1=K base+2*half+1.
    v2f afrag[16];
    const float* arow = x + (size_t)(row0 + l16) * DIM;
#pragma unroll
    for (int kk = 0; kk < 16; ++kk)
        afrag[kk] = *(const v2f*)(arow + 4 * kk + 2 * half);

#pragma unroll
    for (int nt = 0; nt < 4; ++nt) {                 // 16-col output tiles
        const int cb = nt * 16;
        v8f c = {0.f, 0.f, 0.f, 0.f, 0.f, 0.f, 0.f, 0.f};
#pragma unroll
        for (int kk = 0; kk < 16; ++kk) {
            // B layout (4x16 f32): v0 = W[kb+2*half][cb+l16], v1 = W[kb+2*half+1][cb+l16]
            const int r = 4 * kk + 2 * half;
            v2f b;
            b.x = Wl[r * DIM + cb + l16];
            b.y = Wl[(r + 1) * DIM + cb + l16];
            c = __builtin_amdgcn_wmma_f32_16x16x4_f32(
                    false, afrag[kk], false, b, (short)0, c, false, false);
        }
        // D layout: VGPR i -> row row0 + i + 8*half, col cb + l16
#pragma unroll
        for (int i = 0; i < 8; ++i)
            xt[(size_t)(row0 + i + 8 * half) * DIM + cb + l16] = c[i];
    }
}

// ---------- per-node attention scores: s_src = xt.a_src, s_dst = xt.a_dst ----------
__global__ __launch_bounds__(256) void k_node_scores(const float* __restrict__ xt,
                                                     const float* __restrict__ a,  // 192
                                                     float* __restrict__ s_src,
                                                     float* __restrict__ s_dst, int n) {
    const int wave = threadIdx.x >> 5, lane = threadIdx.x & 31;
    const int node = blockIdx.x * 8 + wave;
    if (node >= n) return;
    const float* r = xt + (size_t)node * DIM;
    float v0 = r[lane], v1 = r[lane + 32];
    float ss = v0 * a[lane] + v1 * a[lane + 32];             // a_src = a[0:64]
    float sd = v0 * a[128 + lane] + v1 * a[160 + lane];      // a_dst = a[128:192]
#pragma unroll
    for (int o = 16; o >= 1; o >>= 1) {
        ss += __shfl_xor(ss, o);
        sd += __shfl_xor(sd, o);
    }
    if (lane == 0) { s_src[node] = ss; s_dst[node] = sd; }
}

// ---------- s_rel[r] = (rel_emb[r] @ W_r) . a_rel ----------
__global__ __launch_bounds__(64) void k_rel_scores(const float* __restrict__ rel,  // 32x64
                                                   const float* __restrict__ Wr,   // 64x64
                                                   const float* __restrict__ a_rel,// 64
                                                   float* __restrict__ s_rel) {
    __shared__ float red[DIM];
    const int r = blockIdx.x, j = threadIdx.x;
    float t = 0.f;
#pragma unroll 8
    for (int k = 0; k < DIM; ++k) t += rel[r * DIM + k] * Wr[k * DIM + j];
    red[j] = t * a_rel[j];
    __syncthreads();
    for (int s = 32; s >= 1; s >>= 1) {
        if (j < s) red[j] += red[j + s];
        __syncthreads();
    }
    if (j == 0) s_rel[r] = red[0];
}

// ---------- edge scores + leaky relu + global max ----------
__global__ __launch_bounds__(256) void k_edge_scores(const int* __restrict__ src,
                                                     const int* __restrict__ dst,
                                                     const int* __restrict__ etype,
                                                     const float* __restrict__ s_src,
                                                     const float* __restrict__ s_dst,
                                                     const float* __restrict__ s_rel,
                                                     float* __restrict__ score,
                                                     unsigned* __restrict__ maxbits,
                                                     int n_edges) {
    __shared__ float red[256];
    const int e = blockIdx.x * 256 + threadIdx.x;
    float sc = -3.0e38f;
    if (e < n_edges) {
        float v = s_src[src[e]] + s_rel[etype[e]] + s_dst[dst[e]];
        sc = (v > 0.f) ? v : 0.2f * v;
        score[e] = sc;
    }
    red[threadIdx.x] = sc;
    __syncthreads();
    for (int s = 128; s >= 1; s >>= 1) {
        if (threadIdx.x < s) red[threadIdx.x] = fmaxf(red[threadIdx.x], red[threadIdx.x + s]);
        __syncthreads();
    }
    if (threadIdx.x == 0) atomicMax(maxbits, f2ord(red[0]));
}

// ---------- exp + segment sum of attention ----------
__global__ __launch_bounds__(256) void k_edge_exp(const float* __restrict__ score,
                                                  const int* __restrict__ dst,
                                                  const unsigned* __restrict__ maxbits,
                                                  float* __restrict__ att_exp,
                                                  float* __restrict__ att_sum, int n_edges) {
    const int e = blockIdx.x * 256 + threadIdx.x;
    if (e >= n_edges) return;
    const float m = ord2f(*maxbits);
    const float a = expf(score[e] - m);
    att_exp[e] = a;
    atomicAdd(&att_sum[dst[e]], a);
}

// ---------- message scatter: h[dst] += att_norm * xt[src] (64 lanes / edge) ----------
__global__ __launch_bounds__(256) void k_edge_messages(const int* __restrict__ src,
                                                       const int* __restrict__ dst,
                                                       const float* __restrict__ att_exp,
                                                       const float* __restrict__ att_sum,
                                                       const float* __restrict__ xt,
                                                       float* __restrict__ h, int n_edges) {
    const int t = threadIdx.x;
    const int e = blockIdx.x * 4 + (t >> 6);
    const int k = t & 63;
    if (e >= n_edges) return;
    const int s = src[e], d = dst[e];
    const float w = att_exp[e] / (att_sum[d] + 1e-10f);
    atomicAdd(&h[(size_t)d * DIM + k], w * xt[(size_t)s * DIM + k]);
}

// ---------- ELU + L2 normalize + accumulate ----------
__global__ __launch_bounds__(256) void k_post(const float* __restrict__ h,
                                              float* __restrict__ x,
                                              float* __restrict__ acc, int n) {
    const int wave = threadIdx.x >> 5, lane = threadIdx.x & 31;
    const int node = blockIdx.x * 8 + wave;
    if (node >= n) return;
    float v0 = h[(size_t)node * DIM + lane];
    float v1 = h[(size_t)node * DIM + lane + 32];
    v0 = (v0 > 0.f) ? v0 : (expf(v0) - 1.0f);
    v1 = (v1 > 0.f) ? v1 : (expf(v1) - 1.0f);
    float ss = v0 * v0 + v1 * v1;
#pragma unroll
    for (int o = 16; o >= 1; o >>= 1) ss += __shfl_xor(ss, o);
    const float inv = 1.0f / fmaxf(sqrtf(ss), 1e-12f);
    v0 *= inv; v1 *= inv;
    x[(size_t)node * DIM + lane] = v0;
    x[(size_t)node * DIM + lane + 32] = v1;
    acc[(size_t)node * DIM + lane] += v0;
    acc[(size_t)node * DIM + lane + 32] += v1;
}

__global__ void k_final(const float* __restrict__ acc, float* __restrict__ out, int n) {
    const int i = blockIdx.x * blockDim.x + threadIdx.x;
    if (i < n) out[i] = acc[i] * (1.0f / 3.0f);
}

// ---------- launch ----------
extern "C" void kernel_launch(void* const* d_in, const int* in_sizes, int n_in,
                              void* d_out, int out_size, void* d_ws, size_t ws_size,
                              hipStream_t stream) {
    const float* node_emb = (const float*)d_in[0];          // N*64
    const float* W        = (const float*)d_in[1];          // 2*64*64
    const float* W_r      = (const float*)d_in[2];          // 2*64*64
    const float* a        = (const float*)d_in[3];          // 2*192
    const float* rel_emb  = (const float*)d_in[4];          // 2*32*64
    const int*   edge_idx = (const int*)d_in[5];            // 2*E
    const int*   edge_typ = (const int*)d_in[6];            // E
    const int* src = edge_idx;
    const int* dst = edge_idx + N_EDGES;

    // workspace layout (floats)
    float* xt      = (float*)d_ws;                          // N*64
    float* x       = xt + (size_t)N_NODES * DIM;            // N*64
    float* acc     = x + (size_t)N_NODES * DIM;             // N*64
    float* h       = acc + (size_t)N_NODES * DIM;           // N*64
    float* s_src   = h + (size_t)N_NODES * DIM;             // N
    float* s_dst   = s_src + N_NODES;                       // N
    float* att_sum = s_dst + N_NODES;                       // N
    float* att_exp = att_sum + N_NODES;                     // E
    float* s_rel   = att_exp + N_EDGES;                     // 32
    unsigned* maxbits = (unsigned*)(s_rel + 32);            // 1

    const int nfeat = N_NODES * DIM;
    const dim3 b256(256);

    k_init_xacc<<<(nfeat + 255) / 256, b256, 0, stream>>>(node_emb, x, acc, nfeat);

    for (int l = 0; l < 2; ++l) {
        k_zero<<<(nfeat + 255) / 256, b256, 0, stream>>>(h, nfeat);
        k_zero<<<(N_NODES + 255) / 256, b256, 0, stream>>>(att_sum, N_NODES);
        k_init_max<<<1, 1, 0, stream>>>(maxbits);

        k_gemm_xw<<<(N_NODES / 16 + 7) / 8, b256, 0, stream>>>(x, W + l * DIM * DIM, xt, N_NODES);
        k_node_scores<<<(N_NODES + 7) / 8, b256, 0, stream>>>(xt, a + l * 3 * DIM, s_src, s_dst, N_NODES);
        k_rel_scores<<<32, 64, 0, stream>>>(rel_emb + l * 32 * DIM, W_r + l * DIM * DIM,
                                            a + l * 3 * DIM + DIM, s_rel);
        k_edge_scores<<<(N_EDGES + 255) / 256, b256, 0, stream>>>(src, dst, edge_typ, s_src, s_dst,
                                                                  s_rel, att_exp /*reuse as score*/,
                                                                  maxbits, N_EDGES);
        // att_exp buffer holds raw scores first, then is overwritten with exp in-place
        k_edge_exp<<<(N_EDGES + 255) / 256, b256, 0, stream>>>(att_exp, dst, maxbits, att_exp,
                                                               att_sum, N_EDGES);
        k_edge_messages<<<(N_EDGES + 3) / 4, b256, 0, stream>>>(src, dst, att_exp, att_sum, xt, h,
                                                                N_EDGES);
        k_post<<<(N_NODES + 7) / 8, b256, 0, stream>>>(h, x, acc, N_NODES);
    }

    k_final<<<(nfeat + 255) / 256, b256, 0, stream>>>(acc, (float*)d_out, nfeat);
}